// SparsifyActivation_89335319757222
// MI455X (gfx1250) — compile-verified
//
#include <hip/hip_runtime.h>

#define DIM   4096
#define BLOCK 256
#define VPT   (DIM / BLOCK)   // 16 elements per thread
#define NW    (BLOCK / 32)    // 8 waves per block (wave32)

#if defined(__has_builtin)
# if __has_builtin(__builtin_amdgcn_global_load_async_to_lds_b128)
#  define USE_ASYNC_LDS 1
# endif
#endif

// 4 x i32 vector matching the async-LDS builtin's parameter type
typedef int   v4i __attribute__((__vector_size__(16)));
// clang ext-vector for non-temporal stores
typedef float vf4 __attribute__((ext_vector_type(4)));

__device__ __forceinline__ unsigned key_of(float f) {
    // order-preserving float -> uint map (ascending)
    unsigned u = __float_as_uint(f);
    unsigned mask = (unsigned)(((int)u) >> 31) | 0x80000000u;
    return u ^ mask;
}

__global__ __launch_bounds__(BLOCK)
void topk_sparsify_kernel(const float* __restrict__ x,
                          float* __restrict__ out,
                          int k) {
    __shared__ __align__(16) float sData[DIM];     // 16 KB row staging
    __shared__ unsigned sHistC[NW][256];           // per-wave private histograms (8 KB)
    __shared__ unsigned sHist[BLOCK];              // summed histogram / scan buffer
    __shared__ unsigned sSelBin;
    __shared__ unsigned sKRem;

    const int tid = threadIdx.x;
    const int wid = tid >> 5;                      // wave id (wave32)
    const long long row = blockIdx.x;
    const float* __restrict__ rowIn  = x   + row * (long long)DIM;
    float*       __restrict__ rowOut = out + row * (long long)DIM;

    // ---------------- Phase 0: load row into LDS ----------------
#ifdef USE_ASYNC_LDS
    {
        __attribute__((address_space(1))) v4i* gsrc =
            (__attribute__((address_space(1))) v4i*)rowIn;
        __attribute__((address_space(3))) v4i* ldst =
            (__attribute__((address_space(3))) v4i*)sData;
        #pragma unroll
        for (int i = 0; i < VPT / 4; ++i) {
            int v = tid + i * BLOCK;   // b128 index, coalesced per lane
            __builtin_amdgcn_global_load_async_to_lds_b128(gsrc + v, ldst + v, 0, 0);
        }
# if defined(__has_builtin) && __has_builtin(__builtin_amdgcn_s_wait_asynccnt)
        __builtin_amdgcn_s_wait_asynccnt(0);
# else
        asm volatile("s_wait_asynccnt 0" ::: "memory");
# endif
    }
#else
    #pragma unroll
    for (int i = 0; i < VPT / 4; ++i) {
        int v = tid + i * BLOCK;                    // float4 index, coalesced
        ((float4*)sData)[v] = ((const float4*)rowIn)[v];
    }
#endif
    __syncthreads();

    // ---------------- Phase 1: 4-pass radix select (MSB first) ----------------
    // Finds K = key of the k-th largest element; kRem = #equal-to-K to keep.
    unsigned prefix = 0;
    unsigned kRem   = (unsigned)k;

    #pragma unroll
    for (int pass = 0; pass < 4; ++pass) {
        const int shift = 24 - pass * 8;
        const unsigned hiMask = (pass == 0) ? 0u : (0xFFFFFFFFu << (shift + 8));

        // zero all per-wave histogram copies
        #pragma unroll
        for (int w = 0; w < NW; ++w) sHistC[w][tid] = 0;
        __syncthreads();

        // per-wave privatized histogram: no cross-wave same-address RMW stalls
        #pragma unroll
        for (int i = 0; i < VPT; ++i) {
            unsigned key = key_of(sData[tid + i * BLOCK]);
            if ((key & hiMask) == prefix) {
                atomicAdd(&sHistC[wid][(key >> shift) & 0xFFu], 1u);
            }
        }
        __syncthreads();

        // reduce the 8 copies (conflict-free column sum), stage for scan
        unsigned tot = 0;
        #pragma unroll
        for (int w = 0; w < NW; ++w) tot += sHistC[w][tid];
        sHist[tid] = tot;
        __syncthreads();

        // inclusive SUFFIX scan over 256 bins (Hillis-Steele)
        #pragma unroll
        for (int st = 1; st < BLOCK; st <<= 1) {
            unsigned v = sHist[tid];
            unsigned w = (tid + st < BLOCK) ? sHist[tid + st] : 0u;
            __syncthreads();
            sHist[tid] = v + w;
            __syncthreads();
        }

        // select bin holding the kRem-th largest among matching elements
        unsigned sufb = sHist[tid];
        unsigned sufn = (tid < BLOCK - 1) ? sHist[tid + 1] : 0u;
        if (sufb >= kRem && sufn < kRem) {
            sSelBin = (unsigned)tid;
            sKRem   = kRem - sufn;   // subtract count of strictly-higher bins
        }
        __syncthreads();

        prefix |= (sSelBin << shift);
        kRem    = sKRem;
        __syncthreads();
    }

    const unsigned K = prefix;   // exact 32-bit threshold key

    // ---------------- Phase 2: stable rank of elements equal to K ----------------
    // Thread t owns contiguous indices [t*16, t*16+16) so index order == rank order.
    unsigned eqc = 0;
    #pragma unroll
    for (int j = 0; j < VPT; ++j) {
        eqc += (key_of(sData[tid * VPT + j]) == K) ? 1u : 0u;
    }
    sHist[tid] = eqc;
    __syncthreads();

    // inclusive PREFIX scan over per-thread equal counts
    #pragma unroll
    for (int st = 1; st < BLOCK; st <<= 1) {
        unsigned v = sHist[tid];
        unsigned w = (tid >= st) ? sHist[tid - st] : 0u;
        __syncthreads();
        sHist[tid] = v + w;
        __syncthreads();
    }
    unsigned rank = sHist[tid] - eqc;   // exclusive: # equals at smaller indices

    // ---------------- Phase 3: emit value or zero (non-temporal stores) ----------------
    #pragma unroll
    for (int j4 = 0; j4 < VPT / 4; ++j4) {
        float4 in = ((const float4*)sData)[tid * (VPT / 4) + j4];
        float vv[4] = {in.x, in.y, in.z, in.w};
        float ov[4];
        #pragma unroll
        for (int e = 0; e < 4; ++e) {
            unsigned key = key_of(vv[e]);
            float o = 0.0f;
            if (key > K) {
                o = vv[e];
            } else if (key == K) {
                if (rank < kRem) o = vv[e];   // lowest-index-first tie break
                ++rank;
            }
            ov[e] = o;
        }
        vf4 outv = {ov[0], ov[1], ov[2], ov[3]};
        __builtin_nontemporal_store(outv, (vf4*)rowOut + tid * (VPT / 4) + j4);
    }
}

extern "C" void kernel_launch(void* const* d_in, const int* in_sizes, int n_in,
                              void* d_out, int out_size, void* d_ws, size_t ws_size,
                              hipStream_t stream) {
    (void)n_in; (void)d_ws; (void)ws_size; (void)out_size;
    const float* x = (const float*)d_in[0];
    float* out = (float*)d_out;

    const int dim = DIM;
    const long long rows = (long long)in_sizes[0] / dim;   // 4*4096 = 16384

    int k = (int)((double)dim * (1.0 - 0.95));             // matches Python int(): 204
    if (k < 1) k = 1;

    topk_sparsify_kernel<<<dim3((unsigned)rows), dim3(BLOCK), 0, stream>>>(x, out, k);
}